// GCNModel_38706245272102
// MI455X (gfx1250) — compile-verified
//
#include <hip/hip_runtime.h>
#include <hip/hip_bf16.h>

// ---------------------------------------------------------------------------
// DGCNN EdgeConv pipeline for MI455X (gfx1250, wave32, fp32 WMMA 16x16x4).
//
//   1) rownorm:      sq[i] = ||X[i]||^2
//   2) prep:         W1d = W1[:D]-W1[D:],  W2d/W2b padded 10->16 cols
//   3) knn:          fused (sq_i - 2 X X^T + sq_j) + per-row top-32,
//                    WMMA_F32_16X16X4_F32 tiles, never materializes d2
//   4) gemm x4:      U1=X@W1d, V1=X@W1b, U2=h@W2d, V2=h@W2b   (fp32 WMMA)
//   5) gather-max:   h[i]  = relu(U1[i]+b1+max_j V1[idx]),
//                    out[i]=      U2[i]+b2+max_j V2[idx]
// ---------------------------------------------------------------------------

typedef float v2f __attribute__((ext_vector_type(2)));
typedef float v8f __attribute__((ext_vector_type(8)));

#define NPTS 16384
#define DIMF 128
#define HID  128
#define NCLS 10
#define KNN  32

__device__ __forceinline__ v8f wmma_f32x4(v2f a, v2f b, v8f c) {
  // D = A(16x4) * B(4x16) + C, fp32 everywhere.
  return __builtin_amdgcn_wmma_f32_16x16x4_f32(
      /*neg_a=*/false, a, /*neg_b=*/false, b,
      /*c_mod=*/(short)0, c, /*reuse_a=*/false, /*reuse_b=*/false);
}

// ------------------------------ row norms ----------------------------------
__global__ void dg_rownorm_kernel(const float* __restrict__ X,
                                  float* __restrict__ sq) {
  int i = blockIdx.x * blockDim.x + threadIdx.x;
  if (i >= NPTS) return;
  const float4* xr = (const float4*)(X + (size_t)i * DIMF);
  float s = 0.0f;
#pragma unroll
  for (int t = 0; t < DIMF / 4; ++t) {
    float4 v = xr[t];
    s += v.x * v.x + v.y * v.y + v.z * v.z + v.w * v.w;
  }
  sq[i] = s;
}

// ------------------------------ weight prep --------------------------------
// W1d[k][h] = W1[k][h] - W1[k+128][h]            (128x128)
// W2d[k][c] = W2[k][c] - W2[k+128][c], c<10 else 0   (128x16 padded)
// W2b[k][c] = W2[k+128][c],           c<10 else 0   (128x16 padded)
__global__ void dg_prep_kernel(const float* __restrict__ W1,
                               const float* __restrict__ W2,
                               float* __restrict__ W1d,
                               float* __restrict__ W2d,
                               float* __restrict__ W2b) {
  int t = blockIdx.x * blockDim.x + threadIdx.x;
  if (t < DIMF * HID) {
    W1d[t] = W1[t] - W1[DIMF * HID + t];
  }
  int u = t - DIMF * HID;
  if (u >= 0 && u < DIMF * 16) {
    int k = u >> 4, c = u & 15;
    float dv = 0.0f, bv = 0.0f;
    if (c < NCLS) {
      float top = W2[k * NCLS + c];
      float bot = W2[(k + DIMF) * NCLS + c];
      dv = top - bot;
      bv = bot;
    }
    W2d[u] = dv;
    W2b[u] = bv;
  }
}

// ------------------------------ fused kNN ----------------------------------
// Block = 128 threads (4 waves). Each wave owns a 16-row block; the block
// streams 16-column tiles of X through LDS; each wave chains 32 fp32 WMMAs
// (K=128) per tile, converts to squared distances, and lanes 0-15 maintain a
// per-row 32-entry "replace current max" candidate list in LDS.
__global__ __launch_bounds__(128) void dg_knn_kernel(
    const float* __restrict__ X, const float* __restrict__ sq,
    int* __restrict__ idx_out) {
  __shared__ float tileX[16][DIMF + 4];  // padded: conflict-free b64 reads
  __shared__ float d2s[4][16][17];
  __shared__ float cd[64][KNN];
  __shared__ int   ci[64][KNN];

  const int tid  = threadIdx.x;
  const int w    = tid >> 5;
  const int lane = tid & 31;
  const int lm   = lane & 15;
  const int hi   = lane >> 4;  // 0: K pair {0,1}, 1: K pair {2,3}
  const int r0   = blockIdx.x * 64 + w * 16;

  // A fragments for this wave's 16 rows: 32 steps x float2 (64 VGPRs).
  v2f areg[32];
  {
    const float* xr = X + (size_t)(r0 + lm) * DIMF;
#pragma unroll
    for (int kk = 0; kk < 32; ++kk) {
      areg[kk] = *(const v2f*)(xr + 4 * kk + (hi << 1));
    }
  }
  float sqr[8];
#pragma unroll
  for (int v = 0; v < 8; ++v) sqr[v] = sq[r0 + v + hi * 8];

  // Candidate lists: lane m (<16) of wave w owns row slot s.
  const int s = w * 16 + lm;
  float rowmax   = 3.4e38f;
  int   rowmaxp  = 0;
  if (lane < 16) {
#pragma unroll
    for (int j = 0; j < KNN; ++j) { cd[s][j] = 3.4e38f; ci[s][j] = 0; }
  }
  __syncthreads();

  for (int c0 = 0; c0 < NPTS; c0 += 16) {
    // Cooperative tile load: 16 rows x 128 floats, 8 threads/row x 4 float4.
    {
      const int row = tid >> 3, seg = tid & 7;
      const float4* src = (const float4*)(X + (size_t)(c0 + row) * DIMF + seg * 16);
      float4* dst = (float4*)&tileX[row][seg * 16];
#pragma unroll
      for (int t = 0; t < 4; ++t) dst[t] = src[t];
    }
    __syncthreads();

    const float sqc = sq[c0 + lm];  // L2-resident
    v8f acc = {0.f, 0.f, 0.f, 0.f, 0.f, 0.f, 0.f, 0.f};
#pragma unroll
    for (int kk = 0; kk < 32; ++kk) {
      v2f b = *(const v2f*)&tileX[lm][4 * kk + (hi << 1)];
      acc = wmma_f32x4(areg[kk], b, acc);
    }
    // d2[m][n] = sq_r[m] + sq_c[n] - 2*dot[m][n]
#pragma unroll
    for (int v = 0; v < 8; ++v) {
      d2s[w][v + hi * 8][lm] = sqr[v] + sqc - 2.0f * acc[v];
    }
    __syncthreads();

    if (lane < 16) {
#pragma unroll 1
      for (int n = 0; n < 16; ++n) {
        float d = d2s[w][lm][n];
        if (d < rowmax) {
          cd[s][rowmaxp] = d;
          ci[s][rowmaxp] = c0 + n;
          float m = -3.4e38f; int mp = 0;
#pragma unroll
          for (int j = 0; j < KNN; ++j) {
            float cv = cd[s][j];
            if (cv > m) { m = cv; mp = j; }
          }
          rowmax = m; rowmaxp = mp;
        }
      }
    }
    __syncthreads();
  }

  if (lane < 16) {
#pragma unroll
    for (int j = 0; j < KNN; ++j)
      idx_out[(size_t)(r0 + lm) * KNN + j] = ci[s][j];
  }
}

// ------------------------------ dense GEMM ---------------------------------
// Y[M x Hout] = A[M x 128] @ W[128 x Hout].   One wave per 16x16 tile,
// 32 chained fp32 WMMAs over K=128. Hout in {128, 16}.
__global__ __launch_bounds__(128) void dg_gemm_kernel(
    const float* __restrict__ A, const float* __restrict__ W,
    float* __restrict__ Y, int M, int Hout) {
  const int gwave = blockIdx.x * 4 + (threadIdx.x >> 5);
  const int lane  = threadIdx.x & 31;
  const int lm    = lane & 15;
  const int hi    = lane >> 4;
  const int ct    = gwave % (Hout >> 4);
  const int rt    = gwave / (Hout >> 4);
  const int r0    = rt * 16;
  const int n0    = ct * 16;
  if (r0 >= M) return;  // wave-uniform

  const float* ar = A + (size_t)(r0 + lm) * DIMF;
  v8f acc = {0.f, 0.f, 0.f, 0.f, 0.f, 0.f, 0.f, 0.f};
#pragma unroll
  for (int kk = 0; kk < 32; ++kk) {
    const int ko = 4 * kk + (hi << 1);
    v2f a = *(const v2f*)(ar + ko);
    v2f b;
    b[0] = W[(size_t)ko * Hout + n0 + lm];
    b[1] = W[(size_t)(ko + 1) * Hout + n0 + lm];
    acc = wmma_f32x4(a, b, acc);
  }
#pragma unroll
  for (int v = 0; v < 8; ++v)
    Y[(size_t)(r0 + v + hi * 8) * Hout + n0 + lm] = acc[v];
}

// ------------------------------ gather + max -------------------------------
// out[i][c] = (relu?)( U[i][c] + bias[c] + max_j V[idx[i][j]][c] ), c < Cvalid
__global__ __launch_bounds__(128) void dg_gather_max_kernel(
    const float* __restrict__ U, const float* __restrict__ V,
    const float* __restrict__ bias, const int* __restrict__ idx,
    float* __restrict__ out, int Hdim, int Cvalid, int do_relu) {
  __shared__ int nb[KNN];
  const int i = blockIdx.x;
  const int c = threadIdx.x;
  if (c < KNN) nb[c] = idx[(size_t)i * KNN + c];
  __syncthreads();
  if (c >= Hdim) return;
  float m = -3.4e38f;
#pragma unroll 4
  for (int j = 0; j < KNN; ++j)
    m = fmaxf(m, V[(size_t)nb[j] * Hdim + c]);
  if (c < Cvalid) {
    float r = m + U[(size_t)i * Hdim + c] + bias[c];
    if (do_relu) r = fmaxf(r, 0.0f);
    out[(size_t)i * Cvalid + c] = r;
  }
}

// ------------------------------ launcher -----------------------------------
extern "C" void kernel_launch(void* const* d_in, const int* in_sizes, int n_in,
                              void* d_out, int out_size, void* d_ws, size_t ws_size,
                              hipStream_t stream) {
  const float* X  = (const float*)d_in[0];  // [N,128]
  const float* W1 = (const float*)d_in[1];  // [256,128]
  const float* b1 = (const float*)d_in[2];  // [128]
  const float* W2 = (const float*)d_in[3];  // [256,10]
  const float* b2 = (const float*)d_in[4];  // [10]
  float* out = (float*)d_out;               // [N,10]

  char* ws = (char*)d_ws;
  const size_t OFF_SQ  = 0;
  const size_t OFF_IDX = OFF_SQ  + (size_t)NPTS * 4;             //   64 KB
  const size_t OFF_W1D = OFF_IDX + (size_t)NPTS * KNN * 4;       //  + 2 MB
  const size_t OFF_W2D = OFF_W1D + (size_t)DIMF * HID * 4;
  const size_t OFF_W2B = OFF_W2D + (size_t)DIMF * 16 * 4;
  const size_t OFF_U1  = OFF_W2B + (size_t)DIMF * 16 * 4;
  const size_t OFF_V1  = OFF_U1  + (size_t)NPTS * HID * 4;       //  + 8 MB
  const size_t OFF_H   = OFF_V1  + (size_t)NPTS * HID * 4;       //  + 8 MB
  const size_t OFF_U2  = OFF_H   + (size_t)NPTS * HID * 4;       //  + 8 MB
  const size_t OFF_V2  = OFF_U2  + (size_t)NPTS * 16 * 4;        //  + 1 MB

  float* sq  = (float*)(ws + OFF_SQ);
  int*   nbr = (int*)  (ws + OFF_IDX);
  float* W1d = (float*)(ws + OFF_W1D);
  float* W2d = (float*)(ws + OFF_W2D);
  float* W2b = (float*)(ws + OFF_W2B);
  float* U1  = (float*)(ws + OFF_U1);
  float* V1  = (float*)(ws + OFF_V1);
  float* h   = (float*)(ws + OFF_H);
  float* U2  = (float*)(ws + OFF_U2);
  float* V2  = (float*)(ws + OFF_V2);

  // 1) row norms
  dg_rownorm_kernel<<<(NPTS + 255) / 256, 256, 0, stream>>>(X, sq);
  // 2) weight prep
  dg_prep_kernel<<<(DIMF * HID + DIMF * 16 + 255) / 256, 256, 0, stream>>>(
      W1, W2, W1d, W2d, W2b);
  // 3) fused kNN (64 rows per block)
  dg_knn_kernel<<<NPTS / 64, 128, 0, stream>>>(X, sq, nbr);
  // 4) layer-1 GEMMs:   U1 = X @ W1d,  V1 = X @ W1[128:]
  dg_gemm_kernel<<<(NPTS / 16) * (HID / 16) / 4, 128, 0, stream>>>(
      X, W1d, U1, NPTS, HID);
  dg_gemm_kernel<<<(NPTS / 16) * (HID / 16) / 4, 128, 0, stream>>>(
      X, W1 + DIMF * HID, V1, NPTS, HID);
  // 5) layer-1 gather-max + bias + relu -> h
  dg_gather_max_kernel<<<NPTS, 128, 0, stream>>>(U1, V1, b1, nbr, h,
                                                 HID, HID, 1);
  // 6) layer-2 GEMMs (cols padded to 16)
  dg_gemm_kernel<<<(NPTS / 16) * (16 / 16) / 4, 128, 0, stream>>>(
      h, W2d, U2, NPTS, 16);
  dg_gemm_kernel<<<(NPTS / 16) * (16 / 16) / 4, 128, 0, stream>>>(
      h, W2b, V2, NPTS, 16);
  // 7) layer-2 gather-max + bias -> out  (b2 read guarded to c<10)
  dg_gather_max_kernel<<<NPTS, 128, 0, stream>>>(U2, V2, b2, nbr, out,
                                                 16, NCLS, 0);
  (void)in_sizes; (void)n_in; (void)out_size; (void)ws_size;
}